// GCN_20907900797391
// MI455X (gfx1250) — compile-verified
//
#include <hip/hip_runtime.h>
#include <hip/hip_bf16.h>

typedef __attribute__((ext_vector_type(2))) float v2f;
typedef __attribute__((ext_vector_type(8))) float v8f;

#define HID 64
#define NEG_SLOPE 0.2f
#define EPS_DEN 1e-16f

// ---------- helpers: order-preserving float<->uint encoding for atomicMax ----------
__device__ __forceinline__ unsigned enc_f(float f) {
    unsigned u = __float_as_uint(f);
    return (u & 0x80000000u) ? ~u : (u | 0x80000000u);
}
__device__ __forceinline__ float dec_f(unsigned u) {
    return (u & 0x80000000u) ? __uint_as_float(u & 0x7FFFFFFFu) : __uint_as_float(~u);
}
// enc(-inf): u=0xFF800000 -> ~u
#define ENC_NEG_INF 0x007FFFFFu

// ---------- scalar c = dot(We, a_e), 1 wave ----------
__global__ void dot64_kernel(const float* __restrict__ a, const float* __restrict__ b,
                             float* __restrict__ out) {
    int lane = threadIdx.x & 31;
    float v = a[lane] * b[lane] + a[lane + 32] * b[lane + 32];
    for (int off = 16; off; off >>= 1) v += __shfl_down(v, off);
    if (lane == 0) out[0] = v;
}

// ---------- H = X @ W  (X:[n,64], W:[64,64], H:[n,64]) using V_WMMA_F32_16X16X4_F32 ----
// one wave per 16x16 output tile; K-loop of 16 steps (K=4 each)
__global__ void gemm64_wmma(const float* __restrict__ X, const float* __restrict__ W,
                            float* __restrict__ H, int nrows) {
    int wave = (int)((blockIdx.x * blockDim.x + threadIdx.x) >> 5);
    int lane = threadIdx.x & 31;
    int tileM = wave >> 2;          // 4 col-tiles (64/16)
    int tileN = wave & 3;
    if (tileM * 16 >= nrows) return;   // whole wave exits together (EXEC stays full)

    int m   = lane & 15;
    int hi  = lane >> 4;            // 0 or 1
    int row = tileM * 16 + m;
    int col = tileN * 16 + (lane & 15);

    const float* xrow = X + (size_t)row * HID + 2 * hi;   // A: K = 4*kk + 2*hi + {0,1}
    const float* wcol = W + (size_t)(2 * hi) * HID + col; // B: rows K, col fixed

    v8f c = {};
#pragma unroll
    for (int kk = 0; kk < 16; ++kk) {
        v2f a, b;
        a.x = xrow[kk * 4 + 0];
        a.y = xrow[kk * 4 + 1];
        b.x = wcol[(size_t)(kk * 4) * HID];
        b.y = wcol[(size_t)(kk * 4) * HID + HID];
        c = __builtin_amdgcn_wmma_f32_16x16x4_f32(false, a, false, b, (short)0, c,
                                                  false, false);
    }
    // D layout: VGPR v -> row = 16*tileM + v + 8*hi, col = 16*tileN + (lane&15)
    float* hout = H + (size_t)(tileM * 16 + 8 * hi) * HID + col;
#pragma unroll
    for (int v = 0; v < 8; ++v) hout[(size_t)v * HID] = c[v];
}

// ---------- per-node attention dots: asrc = h . a_s, adst = h . a_d (warp/node) -----
__global__ void node_dots(const float* __restrict__ H, const float* __restrict__ a_s,
                          const float* __restrict__ a_d, float* __restrict__ asrc,
                          float* __restrict__ adst, int n) {
    int w = (int)((blockIdx.x * blockDim.x + threadIdx.x) >> 5);
    int lane = threadIdx.x & 31;
    if (w >= n) return;
    const float* h = H + (size_t)w * HID;
    float s = h[lane] * a_s[lane] + h[lane + 32] * a_s[lane + 32];
    float d = h[lane] * a_d[lane] + h[lane + 32] * a_d[lane + 32];
    for (int off = 16; off; off >>= 1) {
        s += __shfl_down(s, off);
        d += __shfl_down(d, off);
    }
    if (lane == 0) { asrc[w] = s; adst[w] = d; }
}

// ---------- zero acc, init segment-max / denom ----------
__global__ void init_layer(float* __restrict__ acc, unsigned* __restrict__ menc,
                           float* __restrict__ denom, int n) {
    int t = blockIdx.x * blockDim.x + threadIdx.x;
    if (t < n * HID) acc[t] = 0.0f;
    if (t < n) { menc[t] = ENC_NEG_INF; denom[t] = 0.0f; }
}

// ---------- edge pass 1: alpha + leaky relu + segment max ----------
__global__ void edge_pass1(const int* __restrict__ src, const int* __restrict__ dst,
                           const float* __restrict__ ea, const float* __restrict__ asrc,
                           const float* __restrict__ adst, const float* __restrict__ cscal,
                           float* __restrict__ alpha, unsigned* __restrict__ menc, int ne) {
    int e = blockIdx.x * blockDim.x + threadIdx.x;
    if (e >= ne) return;
    int s = src[e], d = dst[e];
    float al = asrc[s] + adst[d] + ea[e] * cscal[0];
    al = (al > 0.0f) ? al : NEG_SLOPE * al;
    alpha[e] = al;
    atomicMax(&menc[d], enc_f(al));
}

// ---------- edge pass 2: ex = exp(alpha - m[dst]); scatter ex*h[src], denom += ex ----
// one wave per edge
__global__ void edge_pass2(const int* __restrict__ src, const int* __restrict__ dst,
                           const float* __restrict__ alpha, const unsigned* __restrict__ menc,
                           const float* __restrict__ H, float* __restrict__ acc,
                           float* __restrict__ denom, int ne) {
    int w = (int)((blockIdx.x * blockDim.x + threadIdx.x) >> 5);
    int lane = threadIdx.x & 31;
    if (w >= ne) return;
    int s = src[w], d = dst[w];
    float ex = expf(alpha[w] - dec_f(menc[d]));
    const float* h = H + (size_t)s * HID;
    float* o = acc + (size_t)d * HID;
    atomicAdd(&o[lane],      ex * h[lane]);
    atomicAdd(&o[lane + 32], ex * h[lane + 32]);
    if (lane == 0) atomicAdd(&denom[d], ex);
}

// ---------- zero pooling buffers ----------
__global__ void zero_pool(float* __restrict__ psum, float* __restrict__ pcnt, int g) {
    int t = blockIdx.x * blockDim.x + threadIdx.x;
    if (t < g * HID) psum[t] = 0.0f;
    if (t < g) pcnt[t] = 0.0f;
}

// ---------- node finish: normalize softmax, add bias, optional relu, optional pool --
__global__ void node_finish(float* __restrict__ acc, const float* __restrict__ denom,
                            const float* __restrict__ bias, int n, int do_relu,
                            const int* __restrict__ batch, float* __restrict__ psum,
                            float* __restrict__ pcnt) {
    int t = blockIdx.x * blockDim.x + threadIdx.x;
    if (t >= n * HID) return;
    int node = t >> 6;
    int c = t & (HID - 1);
    float v = acc[t] / (denom[node] + EPS_DEN) + bias[c];
    if (do_relu) v = fmaxf(v, 0.0f);
    acc[t] = v;
    if (psum != nullptr) {
        int g = batch[node];
        atomicAdd(&psum[(size_t)g * HID + c], v);
        if (c == 0) atomicAdd(&pcnt[g], 1.0f);
    }
}

// ---------- final: out[g] = (psum[g]/max(cnt,1)) . Wlin + blin  (warp/graph) --------
__global__ void final_linear(const float* __restrict__ psum, const float* __restrict__ pcnt,
                             const float* __restrict__ Wlin, const float* __restrict__ blin,
                             float* __restrict__ out, int ng) {
    int w = (int)((blockIdx.x * blockDim.x + threadIdx.x) >> 5);
    int lane = threadIdx.x & 31;
    if (w >= ng) return;
    float cnt = fmaxf(pcnt[w], 1.0f);
    const float* p = psum + (size_t)w * HID;
    float v = p[lane] * Wlin[lane] + p[lane + 32] * Wlin[lane + 32];
    for (int off = 16; off; off >>= 1) v += __shfl_down(v, off);
    if (lane == 0) out[w] = v / cnt + blin[0];
}

extern "C" void kernel_launch(void* const* d_in, const int* in_sizes, int n_in,
                              void* d_out, int out_size, void* d_ws, size_t ws_size,
                              hipStream_t stream) {
    const int N = in_sizes[0] / HID;   // 100000
    const int E = in_sizes[1] / 2;     // 1600000
    const int G = out_size;            // 256

    const float* x         = (const float*)d_in[0];
    const int*   edge_idx  = (const int*)d_in[1];
    const int*   src       = edge_idx;
    const int*   dst       = edge_idx + E;
    const float* edge_attr = (const float*)d_in[2];
    const int*   batch     = (const int*)d_in[3];
    const float* Wlin      = (const float*)d_in[22];
    const float* blin      = (const float*)d_in[23];
    float* out = (float*)d_out;

    // workspace layout
    size_t N64 = (size_t)N * HID;
    float*    hA    = (float*)d_ws;            // projected h       [N,64]
    float*    hB    = hA + N64;                // acc / layer out   [N,64]
    float*    asrc  = hB + N64;                // [N]
    float*    adst  = asrc + N;                // [N]
    unsigned* menc  = (unsigned*)(adst + N);   // [N]
    float*    denom = (float*)(menc + N);      // [N]
    float*    alpha = denom + N;               // [E]
    float*    cscal = alpha + E;               // [1]
    float*    psum  = cscal + 1;               // [G,64]
    float*    pcnt  = psum + (size_t)G * HID;  // [G]

    const int TPB = 256;
    int gemmWaves   = ((N + 15) / 16) * 4;
    int gemmBlocks  = (gemmWaves * 32 + TPB - 1) / TPB;
    int nodeWBlocks = (N * 32 + TPB - 1) / TPB;     // warp per node
    int ncBlocks    = ((int)N64 + TPB - 1) / TPB;   // thread per (node,chan)
    int eBlocks     = (E + TPB - 1) / TPB;          // thread per edge
    long eWarpT     = (long)E * 32;
    int eWBlocks    = (int)((eWarpT + TPB - 1) / TPB); // warp per edge
    int gBlocks     = (G * HID + TPB - 1) / TPB;
    int gWBlocks    = (G * 32 + TPB - 1) / TPB;

    for (int layer = 0; layer < 3; ++layer) {
        int base = 4 + 6 * layer;
        const float* W   = (const float*)d_in[base + 0];
        const float* We  = (const float*)d_in[base + 1];
        const float* a_s = (const float*)d_in[base + 2];
        const float* a_d = (const float*)d_in[base + 3];
        const float* a_e = (const float*)d_in[base + 4];
        const float* bia = (const float*)d_in[base + 5];
        const float* hin = (layer == 0) ? x : hB;

        dot64_kernel<<<1, 32, 0, stream>>>(We, a_e, cscal);
        gemm64_wmma<<<gemmBlocks, TPB, 0, stream>>>(hin, W, hA, N);
        node_dots<<<nodeWBlocks, TPB, 0, stream>>>(hA, a_s, a_d, asrc, adst, N);
        init_layer<<<ncBlocks, TPB, 0, stream>>>(hB, menc, denom, N);
        edge_pass1<<<eBlocks, TPB, 0, stream>>>(src, dst, edge_attr, asrc, adst, cscal,
                                                alpha, menc, E);
        edge_pass2<<<eWBlocks, TPB, 0, stream>>>(src, dst, alpha, menc, hA, hB, denom, E);
        if (layer == 2) {
            zero_pool<<<gBlocks, TPB, 0, stream>>>(psum, pcnt, G);
            node_finish<<<ncBlocks, TPB, 0, stream>>>(hB, denom, bia, N, 0, batch, psum, pcnt);
        } else {
            node_finish<<<ncBlocks, TPB, 0, stream>>>(hB, denom, bia, N, 1, batch,
                                                      nullptr, nullptr);
        }
    }
    final_linear<<<gWBlocks, TPB, 0, stream>>>(psum, pcnt, Wlin, blin, out, G);
}